// PaiNN_59167469470367
// MI455X (gfx1250) — compile-verified
//
#include <hip/hip_runtime.h>
#include <math.h>

#define NNODES 50000
#define NEDGES 400000
#define FDIM   128
#define NRBF   20
#define PI_OVER_CUT (3.14159265358979323846f / 5.0f)

typedef __attribute__((ext_vector_type(2))) float v2f;
typedef __attribute__((ext_vector_type(8))) float v8f;

// D = A(16xK) x B(Kx16) + C  via V_WMMA_F32_16X16X4_F32, fp32-exact.
__device__ __forceinline__ v8f wmma4(v2f a, v2f b, v8f c) {
  return __builtin_amdgcn_wmma_f32_16x16x4_f32(false, a, false, b, (short)0, c,
                                               false, false);
}

// silu via v_exp_f32 + v_rcp_f32 (avoids the IEEE div_scale/div_fixup chain)
__device__ __forceinline__ float silu_f(float x) {
  return x * __builtin_amdgcn_rcpf(1.0f + __expf(-x));
}

// 16x16 output tile of  A(16xK, row-major lda) @ W^T  where W is [out][K] row-major.
// A-frag: lanes 0-15 -> M=lane, K=k,k+1 ; lanes 16-31 -> M=lane-16, K=k+2,k+3.
// B-frag mirrors with N over lanes.  C/D: reg r -> M = r + 8*(lane>=16), N = n0+(lane&15).
__device__ __forceinline__ v8f gemm_tile(const float* __restrict__ A, int lda,
                                         const float* __restrict__ W, int ldw,
                                         int n0, int K, int lane) {
  const int half = (lane >> 4) << 1;   // 0 or 2
  const int mr = lane & 15;
  const float* aRow = A + mr * lda + half;
  const float* wRow = W + (n0 + mr) * ldw + half;
  v8f acc = {0.f, 0.f, 0.f, 0.f, 0.f, 0.f, 0.f, 0.f};
#pragma unroll 8
  for (int k = 0; k < K; k += 4) {
    v2f a = {aRow[k], aRow[k + 1]};
    v2f b = {wRow[k], wRow[k + 1]};
    acc = wmma4(a, b, acc);
  }
  return acc;
}

// ---------------- Kernel 1: embedding gather + phi MLP (compact 256 cols) ----
__global__ __launch_bounds__(256) void k_node_phi(
    const float* __restrict__ emb, const int* __restrict__ z,
    const float* __restrict__ Wp1, const float* __restrict__ bp1,
    const float* __restrict__ Wp2, const float* __restrict__ bp2,
    float* __restrict__ outS, float* __restrict__ outV,
    float* __restrict__ phi) {
  __shared__ float sS[16 * FDIM];
  __shared__ float sT[16 * FDIM];
  const int tid = threadIdx.x;
  const int lane = tid & 31;
  const int wave = tid >> 5;
  const int node0 = blockIdx.x * 16;

  // gather embeddings -> LDS and s accumulator; zero v accumulator
  for (int i = tid; i < 16 * FDIM; i += 256) {
    int m = i >> 7, f = i & 127;
    float val = emb[(size_t)z[node0 + m] * FDIM + f];
    sS[i] = val;
    outS[(size_t)(node0 + m) * FDIM + f] = val;
  }
  for (int i = tid; i < 16 * FDIM * 3; i += 256)
    outV[(size_t)node0 * (FDIM * 3) + i] = 0.0f;
  __syncthreads();

  // stage 1: t = silu(s @ Wp1^T + bp1); one 16-col tile per wave
  {
    int n0 = wave * 16;
    v8f acc = gemm_tile(sS, FDIM, Wp1, FDIM, n0, FDIM, lane);
    int col = n0 + (lane & 15);
    float bias = bp1[col];
    int rbase = (lane >> 4) << 3;
#pragma unroll
    for (int r = 0; r < 8; ++r)
      sT[(rbase + r) * FDIM + col] = silu_f(acc[r] + bias);
  }
  __syncthreads();

  // stage 2: compact phi = t @ Wp2^T + bp2 ; only global cols [0,128) u [256,384)
  // (middle third multiplies v[src]==0 in the reference and is dead)
  for (int t = wave; t < 16; t += 8) {
    int gcol0 = (t < 8) ? t * 16 : 256 + (t - 8) * 16;
    v8f acc = gemm_tile(sT, FDIM, Wp2, FDIM, gcol0, FDIM, lane);
    int col = lane & 15;
    float bias = bp2[gcol0 + col];
    int ccol = t * 16 + col;
    int rbase = (lane >> 4) << 3;
#pragma unroll
    for (int r = 0; r < 8; ++r)
      phi[(size_t)(node0 + rbase + r) * 256 + ccol] = acc[r] + bias;
  }
}

// ---------------- Kernel 2: edge messages (RBF GEMM + gated scatter) --------
__global__ __launch_bounds__(256) void k_edge_msg(
    const float* __restrict__ pos, const int* __restrict__ src,
    const int* __restrict__ dst, const float* __restrict__ Ww,
    const float* __restrict__ bw, const float* __restrict__ phi,
    float* __restrict__ outS, float* __restrict__ outV) {
  __shared__ float sRbf[64 * NRBF];
  __shared__ float sDir[3][64];
  __shared__ int sSrc[64];
  __shared__ int sDst[64];
  const int tid = threadIdx.x;
  const int lane = tid & 31;
  const int wave = tid >> 5;
  const int e0 = blockIdx.x * 64;

  if (tid < 64) {
    int e = e0 + tid;
    int si = src[e], di = dst[e];
    sSrc[tid] = si;
    sDst[tid] = di;
    float rx = pos[di * 3 + 0] - pos[si * 3 + 0];
    float ry = pos[di * 3 + 1] - pos[si * 3 + 1];
    float rz = pos[di * 3 + 2] - pos[si * 3 + 2];
    float d = fmaxf(sqrtf(rx * rx + ry * ry + rz * rz), 1e-9f);
    float inv = 1.0f / d;
    sDir[0][tid] = rx * inv;
    sDir[1][tid] = ry * inv;
    sDir[2][tid] = rz * inv;
    __builtin_prefetch(&phi[(size_t)si * 256], 0, 0);  // global_prefetch_b8
    // rbf_n = sin(n*theta)/d via angle-addition recurrence: one accurate
    // sincos of theta = pi*d/CUTOFF, then 2 fma per term.
    float st, ct;
    __sincosf(PI_OVER_CUT * d, &st, &ct);  // fast path; theta is well-reduced
    float sn = st, cn = ct;
    float* r = &sRbf[tid * NRBF];
    r[0] = sn * inv;
#pragma unroll
    for (int n = 1; n < NRBF; ++n) {
      float sn1 = sn * ct + cn * st;
      float cn1 = cn * ct - sn * st;
      sn = sn1;
      cn = cn1;
      r[n] = sn * inv;
    }
  }
  __syncthreads();

  // 64 tiles: 4 edge-tiles x 16 compact col-tiles; 8 per wave, uniform loops.
  for (int j = wave; j < 64; j += 8) {
    int mt = j >> 4;
    int nt = j & 15;
    int gcol0 = (nt < 8) ? nt * 16 : 256 + (nt - 8) * 16;
    const int half = (lane >> 4) << 1;
    const int mr = lane & 15;
    const float* aRow = sRbf + (mt * 16 + mr) * NRBF + half;
    const float* wRow = Ww + (gcol0 + mr) * NRBF + half;
    v8f acc = {0.f, 0.f, 0.f, 0.f, 0.f, 0.f, 0.f, 0.f};
#pragma unroll
    for (int k = 0; k < NRBF; k += 4) {  // K = 20 -> 5 WMMA steps
      v2f a = {aRow[k], aRow[k + 1]};
      v2f b = {wRow[k], wRow[k + 1]};
      acc = wmma4(a, b, acc);
    }
    int col = lane & 15;
    float bias = bw[gcol0 + col];
    int ccol = nt * 16 + col;
    int rbase = mt * 16 + ((lane >> 4) << 3);
    if (nt < 8) {  // scalar message -> s accumulator
#pragma unroll
      for (int r = 0; r < 8; ++r) {
        int el = rbase + r;
        float val = (acc[r] + bias) * phi[(size_t)sSrc[el] * 256 + ccol];
        atomicAdd(&outS[(size_t)sDst[el] * FDIM + ccol], val);
      }
    } else {  // directional message -> v accumulator (v[src]==0 term dropped)
      int f = ccol - 128;
#pragma unroll
      for (int r = 0; r < 8; ++r) {
        int el = rbase + r;
        float val = (acc[r] + bias) * phi[(size_t)sSrc[el] * 256 + ccol];
        float* vp = &outV[(size_t)sDst[el] * (FDIM * 3) + f * 3];
        atomicAdd(vp + 0, val * sDir[0][el]);
        atomicAdd(vp + 1, val * sDir[1][el]);
        atomicAdd(vp + 2, val * sDir[2][el]);
      }
    }
  }
}

// ---------------- Kernel 3: update block (6 GEMM stages, in-place) ----------
__global__ __launch_bounds__(256) void k_update(
    const float* __restrict__ Wu, const float* __restrict__ bu,
    const float* __restrict__ Wv, const float* __restrict__ bv,
    const float* __restrict__ Wu1, const float* __restrict__ bu1,
    const float* __restrict__ Wu2, const float* __restrict__ bu2,
    float* __restrict__ outS, float* __restrict__ outV) {
  __shared__ float sV[3][16 * FDIM];    // input v (channel-major)
  __shared__ float sU[3][16 * FDIM];    // U = v@Wu^T + bu
  __shared__ float sVV[3][16 * FDIM];   // V = v@Wv^T + bv ; later aliased as a[16][384]
  __shared__ float sS[16 * FDIM];       // s after messages
  __shared__ float sH[16 * 2 * FDIM];   // h = [s, |V|]
  __shared__ float sT2[16 * FDIM];      // silu(h@Wu1^T+bu1)
  __shared__ float sUV[16 * FDIM];      // sum_c U*V
  float* sA = &sVV[0][0];               // a[16][384] reuses V storage

  const int tid = threadIdx.x;
  const int lane = tid & 31;
  const int wave = tid >> 5;
  const int node0 = blockIdx.x * 16;

  for (int i = tid; i < 16 * FDIM; i += 256)
    sS[i] = outS[(size_t)node0 * FDIM + i];
  for (int i = tid; i < 16 * FDIM * 3; i += 256) {
    int m = i / (FDIM * 3), j = i % (FDIM * 3);
    sV[j % 3][m * FDIM + j / 3] = outV[(size_t)node0 * (FDIM * 3) + i];
  }
  __syncthreads();

  // U and V: 2 mats x 3 channels x 8 col-tiles = 48 jobs
  for (int j = wave; j < 48; j += 8) {
    int mat = j / 24;
    int rem = j % 24;
    int ch = rem >> 3;
    int n0 = (rem & 7) * 16;
    const float* W = mat ? Wv : Wu;
    const float* B = mat ? bv : bu;
    v8f acc = gemm_tile(sV[ch], FDIM, W, FDIM, n0, FDIM, lane);
    int col = n0 + (lane & 15);
    float bias = B[col];
    int rbase = (lane >> 4) << 3;
    float* Cm = mat ? sVV[ch] : sU[ch];
#pragma unroll
    for (int r = 0; r < 8; ++r) Cm[(rbase + r) * FDIM + col] = acc[r] + bias;
  }
  __syncthreads();

  // |V| -> h ;  U.V -> sUV  (after this, sVV storage is free for `a`)
  for (int i = tid; i < 16 * FDIM; i += 256) {
    int m = i >> 7, f = i & 127;
    float x = sVV[0][i], y = sVV[1][i], zc = sVV[2][i];
    sH[m * (2 * FDIM) + f] = sS[i];
    sH[m * (2 * FDIM) + FDIM + f] = sqrtf(x * x + y * y + zc * zc);
    sUV[i] = sU[0][i] * x + sU[1][i] * y + sU[2][i] * zc;
  }
  __syncthreads();

  // t2 = silu(h @ Wu1^T + bu1)  (K = 256)
  {
    int n0 = wave * 16;
    v8f acc = gemm_tile(sH, 2 * FDIM, Wu1, 2 * FDIM, n0, 2 * FDIM, lane);
    int col = n0 + (lane & 15);
    float bias = bu1[col];
    int rbase = (lane >> 4) << 3;
#pragma unroll
    for (int r = 0; r < 8; ++r)
      sT2[(rbase + r) * FDIM + col] = silu_f(acc[r] + bias);
  }
  __syncthreads();

  // a = t2 @ Wu2^T + bu2  (24 col-tiles) -> sA (aliases sVV)
  for (int t = wave; t < 24; t += 8) {
    int n0 = t * 16;
    v8f acc = gemm_tile(sT2, FDIM, Wu2, FDIM, n0, FDIM, lane);
    int col = n0 + (lane & 15);
    float bias = bu2[col];
    int rbase = (lane >> 4) << 3;
#pragma unroll
    for (int r = 0; r < 8; ++r)
      sA[(rbase + r) * (3 * FDIM) + col] = acc[r] + bias;
  }
  __syncthreads();

  // s_out = s + a1 + (U.V)*a2 ;  v_out = v + U*a0   (in place)
  for (int i = tid; i < 16 * FDIM; i += 256) {
    int m = i >> 7, f = i & 127;
    float a0 = sA[m * (3 * FDIM) + f];
    float a1 = sA[m * (3 * FDIM) + FDIM + f];
    float a2 = sA[m * (3 * FDIM) + 2 * FDIM + f];
    outS[(size_t)node0 * FDIM + i] = sS[i] + a1 + sUV[i] * a2;
    size_t vb = (size_t)(node0 + m) * (FDIM * 3) + (size_t)f * 3;
#pragma unroll
    for (int c = 0; c < 3; ++c) outV[vb + c] = sV[c][i] + sU[c][i] * a0;
  }
}

extern "C" void kernel_launch(void* const* d_in, const int* in_sizes, int n_in,
                              void* d_out, int out_size, void* d_ws,
                              size_t ws_size, hipStream_t stream) {
  (void)in_sizes; (void)n_in; (void)out_size; (void)ws_size;
  const float* pos = (const float*)d_in[0];
  const float* emb = (const float*)d_in[1];
  const float* Wp1 = (const float*)d_in[2];
  const float* bp1 = (const float*)d_in[3];
  const float* Wp2 = (const float*)d_in[4];
  const float* bp2 = (const float*)d_in[5];
  const float* Ww  = (const float*)d_in[6];
  const float* bw  = (const float*)d_in[7];
  const float* Wu  = (const float*)d_in[8];
  const float* bu  = (const float*)d_in[9];
  const float* Wv  = (const float*)d_in[10];
  const float* bv  = (const float*)d_in[11];
  const float* Wu1 = (const float*)d_in[12];
  const float* bu1 = (const float*)d_in[13];
  const float* Wu2 = (const float*)d_in[14];
  const float* bu2 = (const float*)d_in[15];
  const int* z   = (const int*)d_in[16];
  const int* src = (const int*)d_in[17];
  const int* dst = (const int*)d_in[18];

  float* outS = (float*)d_out;                       // [N,F]
  float* outV = outS + (size_t)NNODES * FDIM;        // [N,F,3]
  float* phi  = (float*)d_ws;                        // [N,256] compact

  k_node_phi<<<NNODES / 16, 256, 0, stream>>>(emb, z, Wp1, bp1, Wp2, bp2,
                                              outS, outV, phi);
  k_edge_msg<<<NEDGES / 64, 256, 0, stream>>>(pos, src, dst, Ww, bw, phi,
                                              outS, outV);
  k_update<<<NNODES / 16, 256, 0, stream>>>(Wu, bu, Wv, bv, Wu1, bu1, Wu2,
                                            bu2, outS, outV);
}